// lossFunc_9036611190820
// MI455X (gfx1250) — compile-verified
//
#include <hip/hip_runtime.h>
#include <cstdint>

typedef float v2f __attribute__((ext_vector_type(2)));
typedef float v8f __attribute__((ext_vector_type(8)));

namespace {
constexpr int kB      = 256;
constexpr int kT      = 200;
constexpr int kQ      = 2048;
constexpr int kTwoQ   = 2 * kQ;              // 4096 floats per batch row
constexpr int kRows   = kB * (kT - 1);       // 50944 loss rows
constexpr int kTile   = 16;                  // rows per wave tile (WMMA M)
constexpr int kTiles  = kRows / kTile;       // 3184 (exact)
constexpr int kWaves  = 2;                   // waves per block
constexpr int kBlocks = kTiles / kWaves;     // 1592 (exact)
constexpr int kSlice  = 128;                 // f32 per row per LDS slice
constexpr int kSlices = kTwoQ / kSlice;      // 32
constexpr int kChunks = kSlice / 4;          // 32 WMMA K-steps per slice
constexpr int kRowBytes      = kTwoQ * 4;    // 16384
constexpr int kSliceRowBytes = kSlice * 4;   // 512
} // namespace

// One wave owns one 16-row tile. batch rows are streamed into LDS with
// CDNA5 async global->LDS b128 copies (double buffered), then fed to
// V_WMMA_F32_16X16X4_F32 whose B matrix encodes [column index j, first-half
// indicator, 0, ..., 0]. After the K loop, D[m,0] = hot column index,
// D[m,1] = correctness flag (both exact in f32). Then gather pred + BCE.
__global__ __launch_bounds__(kWaves * 32) void dkt_loss_main(
    const float* __restrict__ pred, const float* __restrict__ batch,
    float* __restrict__ ws) {
  __shared__ float sbuf[kWaves][2][kTile * kSlice];  // 2 x 8KB per wave
  __shared__ float sD[kWaves][kTile * 2];
  __shared__ float spart[kWaves];

  const int tid  = threadIdx.x;
  const int wv   = tid >> 5;
  const int lane = tid & 31;
  const int n    = lane & 15;   // WMMA: A row / B column / D column
  const int kh   = lane >> 4;   // lane half selects K pair {0,1} vs {2,3}

  const int tile = blockIdx.x * kWaves + wv;
  const int row0 = tile * kTile;

  // Per-lane global base address of each of the 16 batch rows (delta uses
  // batch[:, t+1]); lane's 16B chunk offset folded in.
  uint64_t rb[kTile];
#pragma unroll
  for (int i = 0; i < kTile; ++i) {
    int gr = row0 + i;
    int bb = gr / (kT - 1);
    int tl = gr - bb * (kT - 1);
    rb[i] = (uint64_t)(uintptr_t)batch +
            (uint64_t)(bb * kT + tl + 1) * (uint64_t)kRowBytes +
            (uint64_t)lane * 16u;
  }

  const uint32_t lbase0 =
      (uint32_t)(uintptr_t)(&sbuf[wv][0][0]) + (uint32_t)lane * 16u;
  const uint32_t lbase1 =
      (uint32_t)(uintptr_t)(&sbuf[wv][1][0]) + (uint32_t)lane * 16u;

  // Prologue: async-copy slice 0 into buffer 0 (16 x b128 = 8KB, coalesced).
#pragma unroll
  for (int i = 0; i < kTile; ++i) {
    uint32_t laddr = lbase0 + (uint32_t)(i * kSliceRowBytes);
    uint64_t gaddr = rb[i];
    asm volatile("global_load_async_to_lds_b128 %0, %1, off" ::"v"(laddr),
                 "v"(gaddr)
                 : "memory");
  }

  const float binc = (n == 0) ? 4.0f : 0.0f;  // col-index B lanes advance by K=4
  v8f acc = {0.f, 0.f, 0.f, 0.f, 0.f, 0.f, 0.f, 0.f};

  for (int s = 0; s < kSlices; ++s) {
    if (s + 1 < kSlices) {  // prefetch next slice into other buffer
      const uint32_t lb   = ((s + 1) & 1) ? lbase1 : lbase0;
      const uint64_t soff = (uint64_t)((s + 1) * kSliceRowBytes);
#pragma unroll
      for (int i = 0; i < kTile; ++i) {
        uint32_t laddr = lb + (uint32_t)(i * kSliceRowBytes);
        uint64_t gaddr = rb[i] + soff;
        asm volatile("global_load_async_to_lds_b128 %0, %1, off" ::"v"(laddr),
                     "v"(gaddr)
                     : "memory");
      }
      // 16 newest copies may stay in flight; the current slice's 16 are done.
      asm volatile("s_wait_asynccnt 16" ::: "memory");
    } else {
      asm volatile("s_wait_asynccnt 0" ::: "memory");
    }

    // B matrix for this slice: col0 = global column index j (f32-exact),
    // col1 = (j < Q) indicator (constant per slice since slices don't
    // straddle the halfway point), all other columns 0.
    const int   k0  = s * kSlice;
    const float ind = (s < kSlices / 2) ? 1.0f : 0.0f;
    float b0, b1;
    if (n == 0) {
      b0 = (float)(k0 + 2 * kh);  // B[K=2*kh][0]
      b1 = b0 + 1.0f;             // B[K=2*kh+1][0]
    } else if (n == 1) {
      b0 = ind;
      b1 = ind;
    } else {
      b0 = 0.0f;
      b1 = 0.0f;
    }

    // A operand: lane (M=n) reads its row's K pair {2*kh, 2*kh+1} from LDS.
    const float* aptr = &sbuf[wv][s & 1][n * kSlice + 2 * kh];
#pragma unroll 4
    for (int c = 0; c < kChunks; ++c) {
      v2f A  = *(const v2f*)(aptr + c * 4);  // ds_load_b64, 8B aligned
      v2f Bm = {b0, b1};
      acc = __builtin_amdgcn_wmma_f32_16x16x4_f32(false, A, false, Bm,
                                                  (short)0, acc, false, false);
      b0 += binc;
      b1 += binc;
    }
  }

  // D layout: VGPR d, lanes 0-15 -> row d, col n; lanes 16-31 -> row d+8.
#pragma unroll
  for (int d = 0; d < 8; ++d) {
    if (n < 2) sD[wv][(d + kh * 8) * 2 + n] = acc[d];
  }
  __syncthreads();

  float term = 0.0f;
  if (lane < kTile) {
    float idxf = sD[wv][lane * 2 + 0];  // exact integer in f32
    float af   = sD[wv][lane * 2 + 1];  // 1.0 if correct else 0.0
    int   j    = (int)(idxf + 0.5f);
    bool  corr = af > 0.5f;
    int   col  = corr ? j : (j - kQ);
    int   gr   = row0 + lane;
    int   bb   = gr / (kT - 1);
    int   tl   = gr - bb * (kT - 1);
    float p    = pred[(size_t)(bb * kT + tl) * (size_t)kQ + (size_t)col];
    if (p > 0.0f) term = corr ? logf(p) : log1pf(-p);
  }
#pragma unroll
  for (int off = 16; off >= 1; off >>= 1) term += __shfl_down(term, off);
  if (lane == 0) spart[wv] = term;
  __syncthreads();
  if (tid == 0) {
    float ssum = 0.0f;
    for (int w = 0; w < kWaves; ++w) ssum += spart[w];
    ws[blockIdx.x] = ssum;  // per-block partial of sum(bce)
  }
}

// Deterministic fixed-order final reduction: loss = -sum(partials).
__global__ __launch_bounds__(256) void dkt_loss_reduce(
    const float* __restrict__ ws, float* __restrict__ out, int nitems) {
  __shared__ float sp[256];
  float s = 0.0f;
  for (int i = threadIdx.x; i < nitems; i += 256) s += ws[i];
  sp[threadIdx.x] = s;
  __syncthreads();
  for (int st = 128; st >= 1; st >>= 1) {
    if ((int)threadIdx.x < st) sp[threadIdx.x] += sp[threadIdx.x + st];
    __syncthreads();
  }
  if (threadIdx.x == 0) out[0] = -sp[0];
}

extern "C" void kernel_launch(void* const* d_in, const int* in_sizes, int n_in,
                              void* d_out, int out_size, void* d_ws,
                              size_t ws_size, hipStream_t stream) {
  (void)in_sizes;
  (void)n_in;
  (void)out_size;
  (void)ws_size;
  const float* pred  = (const float*)d_in[0];
  const float* batch = (const float*)d_in[1];
  float*       ws    = (float*)d_ws;

  dkt_loss_main<<<kBlocks, kWaves * 32, 0, stream>>>(pred, batch, ws);
  dkt_loss_reduce<<<1, 256, 0, stream>>>(ws, (float*)d_out, kBlocks);
}